// DTSHLoss_36636071035242
// MI455X (gfx1250) — compile-verified
//
#include <hip/hip_runtime.h>
#include <hip/hip_bf16.h>

#define NROWS 2048
#define BITS  64
#define NCLS  100
#define ALPHA 1.0f
#define LAMBDA 1.0f

typedef __attribute__((ext_vector_type(2))) float v2f;
typedef __attribute__((ext_vector_type(8))) float v8f;

// ---------------------------------------------------------------------------
// Kernel 1: ip = u @ u^T via V_WMMA_F32_16X16X4_F32.
// One wave per 16x16 output tile; 8 waves/block -> block covers 16 rows x 128 cols.
// Grid: (2048/128, 2048/16) = (16, 128).
// ---------------------------------------------------------------------------
__global__ __launch_bounds__(256) void gemm_uut_wmma(const float* __restrict__ u,
                                                     float* __restrict__ ip) {
    const int lane  = threadIdx.x & 31;
    const int wave  = threadIdx.x >> 5;
    const int tileN = blockIdx.x * 8 + wave;   // 0..127
    const int tileM = blockIdx.y;              // 0..127
    const int row_base = tileM * 16;
    const int col_base = tileN * 16;

    // A 16x4 fp32 fragment: lane l holds M = l&15, K = 2*(l>>4) + {0,1}.
    // B 4x16 fragment mirrors it with N = l&15 (B[k][n] = u[col_base+n][k]).
    const int mn    = lane & 15;
    const int khalf = (lane >> 4) << 1;        // 0 or 2

    const float* __restrict__ arow = u + (size_t)(row_base + mn) * BITS;
    const float* __restrict__ brow = u + (size_t)(col_base + mn) * BITS;

    v8f acc = {};
    #pragma unroll
    for (int k = 0; k < BITS; k += 4) {
        v2f a = *(const v2f*)(arow + k + khalf);
        v2f b = *(const v2f*)(brow + k + khalf);
        // 8 args: (neg_a, A, neg_b, B, c_mod, C, reuse_a, reuse_b)
        acc = __builtin_amdgcn_wmma_f32_16x16x4_f32(false, a, false, b,
                                                    (short)0, acc, false, false);
    }

    // C/D layout: VGPR r, lanes 0-15 -> M=r; lanes 16-31 -> M=r+8; N = lane&15.
    const int mBase = row_base + ((lane >> 4) << 3);
    #pragma unroll
    for (int r = 0; r < 8; ++r) {
        ip[(size_t)(mBase + r) * NROWS + col_base + mn] = acc[r];
    }
}

// ---------------------------------------------------------------------------
// Kernel 2: labels from one-hot y; class counts.
// ---------------------------------------------------------------------------
__global__ __launch_bounds__(256) void labels_kernel(const float* __restrict__ y,
                                                     int* __restrict__ lab,
                                                     int* __restrict__ cnt) {
    int i = blockIdx.x * blockDim.x + threadIdx.x;
    if (i < NROWS) {
        int c = 0;
        for (int j = 0; j < NCLS; ++j) {
            if (y[(size_t)i * NCLS + j] > 0.5f) { c = j; break; }
        }
        lab[i] = c;
        atomicAdd(&cnt[c], 1);
    }
}

// ---------------------------------------------------------------------------
// Kernel 3: per-row triplet-style loss. One block per row.
// Gather positive ip values into LDS (n_pos ~ 20), then stream negatives.
// ---------------------------------------------------------------------------
__global__ __launch_bounds__(256) void rowloss_kernel(const float* __restrict__ ip,
                                                      const int* __restrict__ lab,
                                                      float* __restrict__ accum) {
    __shared__ float posv[NROWS];   // worst case all rows positive
    __shared__ int   npos_sh;
    __shared__ float red[256];

    const int i  = blockIdx.x;
    const int li = lab[i];
    if (threadIdx.x == 0) npos_sh = 0;
    __syncthreads();

    const float* __restrict__ row = ip + (size_t)i * NROWS;

    for (int j = threadIdx.x; j < NROWS; j += 256) {
        if (lab[j] == li) {
            int idx = atomicAdd(&npos_sh, 1);
            posv[idx] = row[j];
        }
    }
    __syncthreads();

    const int npos = npos_sh;          // includes i itself (s[i,i] = true)
    const int nneg = NROWS - npos;

    float lsum = 0.0f;
    if (npos > 0 && nneg > 0) {
        for (int j = threadIdx.x; j < NROWS; j += 256) {
            if (lab[j] != li) {
                const float b = row[j];
                for (int p = 0; p < npos; ++p) {
                    float T = posv[p] - b - ALPHA;
                    T = fminf(fmaxf(T, -100.0f), 50.0f);
                    // softplus(T) - T
                    lsum += __logf(1.0f + __expf(T)) - T;
                }
            }
        }
    }

    red[threadIdx.x] = lsum;
    __syncthreads();
    for (int s = 128; s > 0; s >>= 1) {
        if (threadIdx.x < s) red[threadIdx.x] += red[threadIdx.x + s];
        __syncthreads();
    }

    if (threadIdx.x == 0 && npos > 0 && nneg > 0) {
        float rl = red[0] / ((float)npos * (float)nneg);
        atomicAdd(&accum[0], rl);     // total row loss
        atomicAdd(&accum[1], 1.0f);   // valid-row count
    }
}

// ---------------------------------------------------------------------------
// Kernel 4: quantization regularizer sum((u - sign(u))^2).
// ---------------------------------------------------------------------------
__global__ __launch_bounds__(256) void reg_kernel(const float* __restrict__ u,
                                                  float* __restrict__ accum) {
    __shared__ float red[256];
    int i = blockIdx.x * blockDim.x + threadIdx.x;
    float d2 = 0.0f;
    if (i < NROWS * BITS) {
        float v = u[i];
        float s = (v > 0.0f) ? 1.0f : ((v < 0.0f) ? -1.0f : 0.0f);
        float d = v - s;
        d2 = d * d;
    }
    red[threadIdx.x] = d2;
    __syncthreads();
    for (int s = 128; s > 0; s >>= 1) {
        if (threadIdx.x < s) red[threadIdx.x] += red[threadIdx.x + s];
        __syncthreads();
    }
    if (threadIdx.x == 0) atomicAdd(&accum[2], red[0]);
}

// ---------------------------------------------------------------------------
// Kernel 5: finalize scalar.
// ---------------------------------------------------------------------------
__global__ void finalize_kernel(const float* __restrict__ accum,
                                float* __restrict__ out) {
    float tot  = accum[0];
    float cntv = accum[1];
    float regs = accum[2];
    float loss1 = (cntv > 0.0f) ? (tot / fmaxf(cntv, 1.0f)) : 0.0f;
    float loss2 = LAMBDA * regs / (float)(NROWS * BITS);
    out[0] = loss1 + loss2;
}

// ---------------------------------------------------------------------------
extern "C" void kernel_launch(void* const* d_in, const int* in_sizes, int n_in,
                              void* d_out, int out_size, void* d_ws, size_t ws_size,
                              hipStream_t stream) {
    const float* u = (const float*)d_in[0];   // [2048, 64]
    const float* y = (const float*)d_in[1];   // [2048, 100]
    float* out = (float*)d_out;               // scalar

    char* ws = (char*)d_ws;
    float* ip   = (float*)ws;                                   // 16 MB
    size_t off  = (size_t)NROWS * NROWS * sizeof(float);
    int*   lab  = (int*)(ws + off);   off += NROWS * sizeof(int);
    int*   cnt  = (int*)(ws + off);   off += 128 * sizeof(int); // NCLS padded
    float* accum = (float*)(ws + off);                          // [3]

    // zero class counts + accumulators (graph-capture safe)
    hipMemsetAsync(cnt, 0, 128 * sizeof(int) + 4 * sizeof(float), stream);

    dim3 gGrid(NROWS / 128, NROWS / 16);
    gemm_uut_wmma<<<gGrid, 256, 0, stream>>>(u, ip);

    labels_kernel<<<(NROWS + 255) / 256, 256, 0, stream>>>(y, lab, cnt);

    rowloss_kernel<<<NROWS, 256, 0, stream>>>(ip, lab, accum);

    reg_kernel<<<(NROWS * BITS + 255) / 256, 256, 0, stream>>>(u, accum);

    finalize_kernel<<<1, 1, 0, stream>>>(accum, out);
}